// LPAnalysis_3066606649701
// MI455X (gfx1250) — compile-verified
//
#include <hip/hip_runtime.h>

#define ORDER 16
#define FRAME 160
#define NPAD  20                 // padded lag/coeff length (multiple of 4 >= 17)
#define XPS   (NPAD + FRAME + NPAD)   // 20 zeros | 160 samples | 20 zeros = 200
#define WPB   8                  // waves (frames) per block
#define LDSW  (XPS + NPAD + NPAD + FRAME)  // xps, r, a, y  = 400 floats / wave

typedef __attribute__((ext_vector_type(2))) float v2f;
typedef __attribute__((ext_vector_type(8))) float v8f;

__global__ __launch_bounds__(32 * WPB)
void lpc_residual_kernel(const float* __restrict__ pcm,
                         float* __restrict__ out,
                         int nframes)
{
    __shared__ float smem[WPB * LDSW];
    const int lane  = threadIdx.x & 31;
    const int wave  = threadIdx.x >> 5;
    const int frame = blockIdx.x * WPB + wave;
    if (frame >= nframes) return;            // wave-uniform exit

    float* xps   = smem + wave * LDSW;       // [XPS]   zero-padded frame
    float* r_lds = xps + XPS;                // [NPAD]  autocorrelation
    float* a_lds = r_lds + NPAD;             // [NPAD]  LPC coefficients
    float* y_lds = a_lds + NPAD;             // [FRAME] residual staging

    const float* src = pcm + (size_t)frame * FRAME;

    // ---- stage frame into LDS with zero padding on both sides ----
    #pragma unroll
    for (int q = 0; q < 5; ++q)
        xps[NPAD + lane + 32 * q] = src[lane + 32 * q];
    if (lane < NPAD) { xps[lane] = 0.f; xps[NPAD + FRAME + lane] = 0.f; }
    __builtin_amdgcn_wave_barrier();         // wave-private LDS; DS is in-order per wave

    // ---- autocorrelation: lane l computes r[l] ----
    if (lane <= ORDER) {
        float s = 0.f;
        for (int t = 0; t < FRAME; ++t)
            s += xps[NPAD + t] * xps[NPAD + t + lane];
        r_lds[lane] = s;
    } else if (lane < NPAD) {
        r_lds[lane] = 0.f;
    }
    if (lane == 0) a_lds[0] = 1.f;
    if (lane >= 1 && lane < NPAD) a_lds[lane] = 0.f;
    __builtin_amdgcn_wave_barrier();

    // ---- Levinson-Durbin: uniform values computed redundantly on all lanes;
    //      coefficient update parallel over lanes (lane j owns a[j]).
    //      Plain LDS accesses: per-wave DS ops are in order, the stores' data
    //      depends on the loads, and the runtime-indexed store aliases every
    //      later load, so no volatile is needed (avoids FLAT/SYS lowering).
    {
        float e = r_lds[0];
        for (int i = 1; i <= ORDER; ++i) {
            float acc = r_lds[i];
            for (int j = 1; j < i; ++j)
                acc -= a_lds[j] * r_lds[i - j];
            float k = acc / e;               // uniform across the wave
            const int  j   = lane;
            const bool upd = (j >= 1) && (j < i);
            float aj  = upd ? a_lds[j]     : 0.f;
            float arv = upd ? a_lds[i - j] : 0.f;   // old values read before any write
            float na  = aj - k * arv;
            if (upd)       a_lds[j] = na;
            if (lane == 0) a_lds[i] = k;
            e = e * (1.f - k * k);
            __builtin_amdgcn_wave_barrier();
        }
    }

    // ---- residual FIR via V_WMMA_F32_16X16X4_F32 ----
    // y[t] = sum_{j=0..16} a[j] * xpad[t - j],  a[17..19] == 0.
    // D(16x16) per chunk c: rows m = t - 16c, all 16 columns identical
    // (B[k,n] = a[kk] broadcast across n).  K tiled 5 x 4 over j.
    const int half = lane >> 4;              // A/B K-halves: lanes 0-15 -> K 0,1 ; 16-31 -> K 2,3
    const int mrow = lane & 15;
    const int koff = half * 2;

    v2f bop[5];
    #pragma unroll
    for (int s = 0; s < 5; ++s) {
        bop[s].x = a_lds[4 * s + koff];
        bop[s].y = a_lds[4 * s + koff + 1];
    }

    #pragma unroll
    for (int c = 0; c < 10; ++c) {
        v8f acc = {};
        const int tb = 16 * c;
        #pragma unroll
        for (int s = 0; s < 5; ++s) {
            const int kk = 4 * s + koff;
            v2f aop;
            aop.x = xps[NPAD + tb + mrow - kk];        // A[m, kk]   = xpad[tb+m-kk]
            aop.y = xps[NPAD + tb + mrow - kk - 1];    // A[m, kk+1]
            acc = __builtin_amdgcn_wmma_f32_16x16x4_f32(
                      false, aop, false, bop[s], (short)0, acc, false, false);
        }
        // D layout: lanes 0-15 hold M = vgpr idx (0..7), lanes 16-31 hold M = 8 + idx.
        // All columns identical -> every lane writes the same value (no divergence).
        #pragma unroll
        for (int q = 0; q < 8; ++q)
            y_lds[tb + half * 8 + q] = acc[q];
    }
    __builtin_amdgcn_wave_barrier();

    // ---- coalesced store ----
    float* dst = out + (size_t)frame * FRAME;
    #pragma unroll
    for (int q = 0; q < 5; ++q)
        dst[lane + 32 * q] = y_lds[lane + 32 * q];
}

extern "C" void kernel_launch(void* const* d_in, const int* in_sizes, int n_in,
                              void* d_out, int out_size, void* d_ws, size_t ws_size,
                              hipStream_t stream) {
    const float* pcm = (const float*)d_in[0];
    float* out = (float*)d_out;
    const int total   = in_sizes[0];        // 32 * 160000
    const int nframes = total / FRAME;      // 32000
    const int blocks  = (nframes + WPB - 1) / WPB;
    lpc_residual_kernel<<<blocks, 32 * WPB, 0, stream>>>(pcm, out, nframes);
}